// AttentionModule_38122129719346
// MI455X (gfx1250) — compile-verified
//
#include <hip/hip_runtime.h>
#include <hip/hip_bf16.h>
#include <math.h>

// ---------------- problem constants (fixed by the reference) ----------------
#define BB   4      // batch
#define CC   512    // channels
#define MID  256    // mid = C/2
#define CMM  76     // num_classes * num_local_prototype
#define HH   128
#define WW   128
#define NP   16     // BH*BW patches
#define PP   1024   // pixels per patch (32x32)
#define BN_EPS 1e-5f

typedef float v2f __attribute__((ext_vector_type(2)));
typedef float v8f __attribute__((ext_vector_type(8)));

// ------------------------- WMMA f32 16x16x4 helpers -------------------------
// A: 16x4 f32, lane L holds A[L%16][k + 2*(L/16)] in v.x, +1 in v.y
// B: 4x16 f32, lane L holds B[k + 2*(L/16)][n0 + L%16] in v.x, next K row in v.y
// C/D: 8 VGPRs; VGPR r -> C[m0 + r + 8*(L/16)][n0 + L%16]
__device__ __forceinline__ v8f wmma4(v2f a, v2f b, v8f c) {
  return __builtin_amdgcn_wmma_f32_16x16x4_f32(false, a, false, b, (short)0, c,
                                               false, false);
}

__device__ __forceinline__ v2f fragA(const float* A, int lda, int m0, int kk,
                                     int lane) {
  int r  = m0 + (lane & 15);
  int kb = kk + ((lane >> 4) << 1);
  v2f a;
  a.x = A[r * lda + kb];
  a.y = A[r * lda + kb + 1];
  return a;
}

__device__ __forceinline__ v2f fragB(const float* Bm, int ldb, int n0, int kk,
                                     int lane) {
  int c  = n0 + (lane & 15);
  int kb = kk + ((lane >> 4) << 1);
  v2f b;
  b.x = Bm[kb * ldb + c];
  b.y = Bm[(kb + 1) * ldb + c];
  return b;
}

// ---------------- kernel 1: BN affine precompute + v-pad zero ---------------
__global__ __launch_bounds__(256) void k_bnprep(const float* gamma,
                                                const float* beta,
                                                const float* mean,
                                                const float* var,
                                                float* bns, float* bnb,
                                                float* vmat) {
  int tid = threadIdx.x;
  for (int i = tid; i < CC; i += 256) {
    float inv = gamma[i] * rsqrtf(var[i] + BN_EPS);
    bns[i] = inv;
    bnb[i] = beta[i] - mean[i] * inv;
  }
  // zero pad rows 76..79 of v (K padded to 80 in the attention kernel)
  for (int i = tid; i < 4 * MID; i += 256) vmat[CMM * MID + i] = 0.f;
}

// ------------- kernel 2: patch means -> pcs output + conf workspace ---------
__global__ __launch_bounds__(256) void k_pcs(const float* sim, float* pcs_out,
                                             float* conf) {
  __shared__ float red[256];
  int idx = blockIdx.x;            // (b*CM + cm)*16 + n
  int n   = idx & 15;
  int cm  = (idx >> 4) % CMM;
  int b   = idx / (16 * CMM);
  int tid = threadIdx.x;
  const float* sp = sim + (size_t)(b * CMM + cm) * HH * WW;
  int Rb = (n >> 2) * 32, Cb = (n & 3) * 32;
  float s = 0.f;
#pragma unroll
  for (int j = 0; j < 4; j++) {
    int p = tid + 256 * j;
    s += sp[(size_t)(Rb + (p >> 5)) * WW + Cb + (p & 31)];
  }
  red[tid] = s;
  __syncthreads();
  for (int off = 128; off > 0; off >>= 1) {
    if (tid < off) red[tid] += red[tid + off];
    __syncthreads();
  }
  if (tid == 0) {
    float m = red[0] * (1.f / 1024.f);
    pcs_out[(size_t)(b * CMM + cm) * 16 + n] = m;
    conf[(size_t)(b * 16 + n) * CMM + cm]    = m;
  }
}

// ------- kernel 3: softmax over P (conf folded in) -> S' [bn][cm][p] --------
__global__ __launch_bounds__(256) void k_softmax(const float* sim,
                                                 const float* conf, float* Sp) {
  __shared__ float red[256];
  int idx = blockIdx.x;            // bn*CM + cm
  int cm  = idx % CMM;
  int bn  = idx / CMM;
  int b = bn >> 4, n = bn & 15;
  int tid = threadIdx.x;
  const float* sp = sim + (size_t)(b * CMM + cm) * HH * WW;
  int Rb = (n >> 2) * 32, Cb = (n & 3) * 32;
  float v[4];
  float lm = -INFINITY;
#pragma unroll
  for (int j = 0; j < 4; j++) {
    int p = tid + 256 * j;
    v[j] = sp[(size_t)(Rb + (p >> 5)) * WW + Cb + (p & 31)];
    lm   = fmaxf(lm, v[j]);
  }
  red[tid] = lm;
  __syncthreads();
  for (int off = 128; off > 0; off >>= 1) {
    if (tid < off) red[tid] = fmaxf(red[tid], red[tid + off]);
    __syncthreads();
  }
  float mx = red[0];
  __syncthreads();
  float ls = 0.f;
#pragma unroll
  for (int j = 0; j < 4; j++) ls += __expf(v[j] - mx);
  red[tid] = ls;
  __syncthreads();
  for (int off = 128; off > 0; off >>= 1) {
    if (tid < off) red[tid] += red[tid + off];
    __syncthreads();
  }
  float scale = conf[(size_t)bn * CMM + cm] / red[0];
  float* out = Sp + (size_t)idx * PP;
#pragma unroll
  for (int j = 0; j < 4; j++) {
    int p = tid + 256 * j;
    out[p] = __expf(v[j] - mx) * scale;
  }
}

// ---------- kernel 4: local[bn][cm][c] = S'[cm][p] @ xs[p][c] (K=1024) ------
__global__ __launch_bounds__(256) void k_gemm_local(const float* Sp,
                                                    const float* x,
                                                    float* localw) {
  __shared__ float As[32 * 34];
  __shared__ float Bs[32 * 66];
  int bn = blockIdx.z;
  int b = bn >> 4, n = bn & 15;
  int m0 = blockIdx.y * 32;        // cm tile (3 tiles, guard < 76)
  int c0 = blockIdx.x * 64;        // channel tile
  int tid = threadIdx.x, wave = tid >> 5, lane = tid & 31;
  int wm = (wave >> 2) * 16, wn = (wave & 3) * 16;
  int Rbase = (n >> 2) * 32, Cbase = (n & 3) * 32;
  const float* Abn = Sp + (size_t)bn * CMM * PP;
  v8f acc = {};
  for (int kt = 0; kt < PP; kt += 32) {
    __syncthreads();
    {  // stage A (S'): 32 rows x 32 K
      int r = tid >> 3, cc = (tid & 7) * 4, mr = m0 + r;
#pragma unroll
      for (int j = 0; j < 4; j++) {
        float vv = 0.f;
        if (mr < CMM) vv = Abn[(size_t)mr * PP + kt + cc + j];
        As[r * 34 + cc + j] = vv;
      }
    }
    {  // stage B (xs): K-tile = one pixel row of the patch -> coalesced
      int pl = tid & 31;
      int R  = Rbase + (kt >> 5);
      int Cw = Cbase + pl;
#pragma unroll
      for (int j = 0; j < 8; j++) {
        int cl = (tid >> 5) * 8 + j;
        Bs[pl * 66 + cl] =
            x[(((size_t)b * CC + c0 + cl) * HH + R) * WW + Cw];
      }
    }
    __syncthreads();
#pragma unroll
    for (int k2 = 0; k2 < 32; k2 += 4) {
      v2f a  = fragA(As, 34, wm, k2, lane);
      v2f bb = fragB(Bs, 66, wn, k2, lane);
      acc    = wmma4(a, bb, acc);
    }
  }
#pragma unroll
  for (int r = 0; r < 8; r++) {
    int mr = m0 + wm + r + ((lane >> 4) << 3);
    int ch = c0 + wn + (lane & 15);
    if (mr < CMM) localw[((size_t)bn * CMM + mr) * CC + ch] = acc[r];
  }
}

// ------ kernel 5: batched K (z<64) and V (z==64) projections (K=512) --------
__global__ __launch_bounds__(256) void k_gemm_kv(const float* localw,
                                                 const float* g,
                                                 const float* Wk,
                                                 const float* bk,
                                                 const float* Wv,
                                                 const float* bv, float* kmat,
                                                 float* vmat) {
  __shared__ float As[32 * 34];
  __shared__ float Bs[32 * 66];
  int z = blockIdx.z;
  const float *A, *Bm, *bias;
  float* Out;
  if (z < 64) {
    A = localw + (size_t)z * CMM * CC; Bm = Wk; bias = bk;
    Out = kmat + (size_t)z * CMM * MID;
  } else {
    A = g; Bm = Wv; bias = bv; Out = vmat;
  }
  int m0 = blockIdx.y * 32;
  int c0 = blockIdx.x * 64;
  int tid = threadIdx.x, wave = tid >> 5, lane = tid & 31;
  int wm = (wave >> 2) * 16, wn = (wave & 3) * 16;
  v8f acc = {};
  for (int kt = 0; kt < CC; kt += 32) {
    __syncthreads();
    {
      int r = tid >> 3, ccl = (tid & 7) * 4, mr = m0 + r;
#pragma unroll
      for (int j = 0; j < 4; j++) {
        float vv = 0.f;
        if (mr < CMM) vv = A[(size_t)mr * CC + kt + ccl + j];
        As[r * 34 + ccl + j] = vv;
      }
    }
    {
      int kl = tid >> 3;
#pragma unroll
      for (int j = 0; j < 8; j++) {
        int cl = (tid & 7) * 8 + j;
        Bs[kl * 66 + cl] = Bm[(size_t)(kt + kl) * MID + c0 + cl];
      }
    }
    __syncthreads();
#pragma unroll
    for (int k2 = 0; k2 < 32; k2 += 4) {
      v2f a  = fragA(As, 34, wm, k2, lane);
      v2f bb = fragB(Bs, 66, wn, k2, lane);
      acc    = wmma4(a, bb, acc);
    }
  }
#pragma unroll
  for (int r = 0; r < 8; r++) {
    int mr = m0 + wm + r + ((lane >> 4) << 3);
    int ch = c0 + wn + (lane & 15);
    if (mr < CMM) Out[(size_t)mr * MID + ch] = acc[r] + bias[ch];
  }
}

// --- kernel 6: fused Q-proj -> logits -> softmax -> att@v, per 32-row tile --
__global__ __launch_bounds__(256) void k_attn(const float* x, const float* Wq,
                                              const float* bq,
                                              const float* kmat,
                                              const float* vmat, float* outw) {
  __shared__ float As[32 * 34];
  __shared__ float Bs[32 * 66];
  __shared__ float Qt[32 * 258];   // q tile [32][256], padded
  __shared__ float At[32 * 84];    // logits/att [32][80], padded
  int bn = blockIdx.z;
  int b = bn >> 4, n = bn & 15;
  int p0 = blockIdx.x * 32;
  int tid = threadIdx.x, wave = tid >> 5, lane = tid & 31;
  int wm = (wave >> 2) * 16, wn = (wave & 3) * 16;
  int Rbase = (n >> 2) * 32, Cbase = (n & 3) * 32;
  int R = Rbase + (p0 >> 5);       // all 32 rows = one pixel row of the patch

  // ---- stage 1: q = xs @ Wq + bq  (M=32, N=256, K=512) ----
  v8f qacc[4] = {};
  for (int kt = 0; kt < CC; kt += 32) {
    __syncthreads();
    {  // stage A: xs rows -> transpose-load, coalesced along pixel column
      int pl = tid & 31;
      int Cw = Cbase + pl;
#pragma unroll
      for (int j = 0; j < 4; j++) {
        int cl = (tid >> 5) * 4 + j;
        As[pl * 34 + cl] =
            x[(((size_t)b * CC + kt + cl) * HH + R) * WW + Cw];
      }
    }
    for (int nt = 0; nt < 4; nt++) {
      __syncthreads();
      {
        int kl = tid >> 3;
#pragma unroll
        for (int j = 0; j < 8; j++) {
          int cl = (tid & 7) * 8 + j;
          Bs[kl * 66 + cl] = Wq[(size_t)(kt + kl) * MID + nt * 64 + cl];
        }
      }
      __syncthreads();
#pragma unroll
      for (int k2 = 0; k2 < 32; k2 += 4) {
        v2f a  = fragA(As, 34, wm, k2, lane);
        v2f bb = fragB(Bs, 66, wn, k2, lane);
        qacc[nt] = wmma4(a, bb, qacc[nt]);
      }
    }
  }
  __syncthreads();
#pragma unroll
  for (int nt = 0; nt < 4; nt++)
#pragma unroll
    for (int r = 0; r < 8; r++) {
      int row = wm + r + ((lane >> 4) << 3);
      int col = nt * 64 + wn + (lane & 15);
      Qt[row * 258 + col] = qacc[nt][r] + bq[col];
    }
  __syncthreads();

  // ---- stage 2: logits = q @ k^T  (M=32, N=80(76), K=256) ----
  const float* kp = kmat + (size_t)bn * CMM * MID;
  for (int task = wave; task < 10; task += 8) {
    int mt = task / 5, ntile = task % 5;
    v8f lacc = {};
    for (int k2 = 0; k2 < MID; k2 += 4) {
      v2f a  = fragA(Qt, 258, mt * 16, k2, lane);
      int cm = ntile * 16 + (lane & 15);
      int kb = k2 + ((lane >> 4) << 1);
      v2f bb;
      bb.x = (cm < CMM) ? kp[(size_t)cm * MID + kb] : 0.f;
      bb.y = (cm < CMM) ? kp[(size_t)cm * MID + kb + 1] : 0.f;
      lacc = wmma4(a, bb, lacc);
    }
#pragma unroll
    for (int r = 0; r < 8; r++) {
      int row = mt * 16 + r + ((lane >> 4) << 3);
      int cm  = ntile * 16 + (lane & 15);
      if (cm < CMM) At[row * 84 + cm] = lacc[r];
    }
  }
  __syncthreads();

  // ---- softmax over cm (one thread per row) ----
  if (tid < 32) {
    float mx = -INFINITY;
    for (int c = 0; c < CMM; c++) mx = fmaxf(mx, At[tid * 84 + c]);
    float s = 0.f;
    for (int c = 0; c < CMM; c++) s += __expf(At[tid * 84 + c] - mx);
    float inv = 1.f / s;
    for (int c = 0; c < CMM; c++)
      At[tid * 84 + c] = __expf(At[tid * 84 + c] - mx) * inv;
    for (int c = CMM; c < 80; c++) At[tid * 84 + c] = 0.f;
  }
  __syncthreads();

  // ---- stage 3: out = att @ v  (M=32, N=256, K=80) ----
  float* ob = outw + ((size_t)bn * PP + p0) * MID;
  for (int nt = 0; nt < 4; nt++) {
    v8f oacc = {};
    for (int k2 = 0; k2 < 80; k2 += 4) {
      v2f a    = fragA(At, 84, wm, k2, lane);
      int colg = nt * 64 + wn + (lane & 15);
      int kb   = k2 + ((lane >> 4) << 1);
      v2f bb;
      bb.x = vmat[(size_t)kb * MID + colg];
      bb.y = vmat[(size_t)(kb + 1) * MID + colg];
      oacc = wmma4(a, bb, oacc);
    }
#pragma unroll
    for (int r = 0; r < 8; r++) {
      int row = wm + r + ((lane >> 4) << 3);
      int col = nt * 64 + wn + (lane & 15);
      ob[(size_t)row * MID + col] = oacc[r];
    }
  }
}

// ---- kernel 7: y = relu(BN(out @ Wc)) with patch-recover scatter (K=256) ---
__global__ __launch_bounds__(256) void k_conv(const float* outw,
                                              const float* Wc,
                                              const float* bns,
                                              const float* bnb, float* y) {
  __shared__ float As[32 * 34];
  __shared__ float Bs[32 * 66];
  int bn = blockIdx.z;
  int b = bn >> 4, n = bn & 15;
  int m0 = blockIdx.y * 32;        // pixel-row tile
  int c0 = blockIdx.x * 64;        // out-channel tile
  int tid = threadIdx.x, wave = tid >> 5, lane = tid & 31;
  int wm = (wave >> 2) * 16, wn = (wave & 3) * 16;
  int Rbase = (n >> 2) * 32, Cbase = (n & 3) * 32;
  const float* A = outw + ((size_t)bn * PP + m0) * MID;
  v8f acc = {};
  for (int kt = 0; kt < MID; kt += 32) {
    __syncthreads();
    {
      int r = tid >> 3, ccl = (tid & 7) * 4;
#pragma unroll
      for (int j = 0; j < 4; j++)
        As[r * 34 + ccl + j] = A[(size_t)r * MID + kt + ccl + j];
    }
    {
      int kl = tid >> 3;
#pragma unroll
      for (int j = 0; j < 8; j++) {
        int cl = (tid & 7) * 8 + j;
        Bs[kl * 66 + cl] = Wc[(size_t)(kt + kl) * CC + c0 + cl];
      }
    }
    __syncthreads();
#pragma unroll
    for (int k2 = 0; k2 < 32; k2 += 4) {
      v2f a  = fragA(As, 34, wm, k2, lane);
      v2f bb = fragB(Bs, 66, wn, k2, lane);
      acc    = wmma4(a, bb, acc);
    }
  }
#pragma unroll
  for (int r = 0; r < 8; r++) {
    int p  = m0 + wm + r + ((lane >> 4) << 3);
    int ch = c0 + wn + (lane & 15);
    float val = fmaxf(acc[r] * bns[ch] + bnb[ch], 0.f);
    int Rr = Rbase + (p >> 5);
    int Cw = Cbase + (p & 31);
    y[(((size_t)b * CC + ch) * HH + Rr) * WW + Cw] = val;
  }
}

// --------------------------------- launcher ---------------------------------
extern "C" void kernel_launch(void* const* d_in, const int* in_sizes, int n_in,
                              void* d_out, int out_size, void* d_ws,
                              size_t ws_size, hipStream_t stream) {
  const float* x    = (const float*)d_in[0];
  const float* sim  = (const float*)d_in[1];
  const float* gctr = (const float*)d_in[2];   // [76,512]
  const float* Wq   = (const float*)d_in[3];
  const float* bq   = (const float*)d_in[4];
  const float* Wk   = (const float*)d_in[5];
  const float* bk   = (const float*)d_in[6];
  const float* Wv   = (const float*)d_in[7];
  const float* bv   = (const float*)d_in[8];
  const float* Wc   = (const float*)d_in[9];
  const float* bng  = (const float*)d_in[10];
  const float* bnb_ = (const float*)d_in[11];
  const float* bnm  = (const float*)d_in[12];
  const float* bnv  = (const float*)d_in[13];

  float* y       = (float*)d_out;                          // [4,512,128,128]
  float* pcs_out = y + (size_t)BB * CC * HH * WW;          // [4,76,4,4]

  // workspace layout (floats); total ~25.5M floats (~102 MB)
  float* ws = (float*)d_ws;
  size_t off = 0;
  float* conf   = ws + off; off += (size_t)BB * NP * CMM;        // 4864
  float* bns    = ws + off; off += CC;                           // 512
  float* bnbuf  = ws + off; off += CC;                           // 512
  float* Sp     = ws + off; off += (size_t)BB * NP * CMM * PP;   // 4.98M
  float* localw = ws + off; off += (size_t)BB * NP * CMM * CC;   // 2.49M
  float* kmat   = ws + off; off += (size_t)BB * NP * CMM * MID;  // 1.25M
  float* vmat   = ws + off; off += (size_t)80 * MID;             // padded to 80
  float* outw   = ws + off; off += (size_t)BB * NP * PP * MID;   // 16.8M
  (void)off; (void)ws_size; (void)in_sizes; (void)n_in; (void)out_size;

  k_bnprep<<<1, 256, 0, stream>>>(bng, bnb_, bnm, bnv, bns, bnbuf, vmat);
  k_pcs<<<BB * CMM * NP, 256, 0, stream>>>(sim, pcs_out, conf);
  k_softmax<<<BB * NP * CMM, 256, 0, stream>>>(sim, conf, Sp);
  k_gemm_local<<<dim3(CC / 64, 3, BB * NP), 256, 0, stream>>>(Sp, x, localw);
  k_gemm_kv<<<dim3(MID / 64, 3, BB * NP + 1), 256, 0, stream>>>(
      localw, gctr, Wk, bk, Wv, bv, kmat, vmat);
  k_attn<<<dim3(PP / 32, 1, BB * NP), 256, 0, stream>>>(x, Wq, bq, kmat, vmat,
                                                        outw);
  k_conv<<<dim3(CC / 64, PP / 32, BB * NP), 256, 0, stream>>>(outw, Wc, bns,
                                                              bnbuf, y);
}